// GCN_84559316124289
// MI455X (gfx1250) — compile-verified
//
#include <hip/hip_runtime.h>

// f32 WMMA operand types: A/B = 16x4 / 4x16 f32 -> 2 VGPRs; C/D = 16x16 f32 -> 8 VGPRs
typedef __attribute__((ext_vector_type(2))) float v2f;
typedef __attribute__((ext_vector_type(8))) float v8f;

// Repack OIHW weights into WMMA-native, tap-major layout with scale folded in:
//   wrep[(o/16)*K + (kh*3+kw)*CIN + cin][o%16] = w[o][cin][kh][kw] * wscale
// so B-operand loads are coalesced (16 lanes read 16 consecutive floats).
__global__ __launch_bounds__(256)
void repack_w(const float* __restrict__ w, float* __restrict__ wrep,
              int COUT, int CIN, float wscale)
{
    const int K   = CIN * 9;
    const int idx = blockIdx.x * 256 + threadIdx.x;
    if (idx >= COUT * K) return;
    const int o   = idx / K;
    const int r   = idx - o * K;      // r = cin*9 + p
    const int cin = r / 9;
    const int p   = r - cin * 9;      // filter tap kh*3+kw
    const int kk  = p * CIN + cin;    // tap-major K index
    wrep[((size_t)(o >> 4) * K + kk) * 16 + (o & 15)] = w[idx] * wscale;
}

// Fused 3x3 SAME conv (implicit GEMM on V_WMMA_F32_16X16X4_F32) + LeakyReLU(0.2)
// + 2x2 maxpool.  One wave computes a 16(pixels: 2 rows x 8 cols) x 16(channels)
// conv tile and emits the 1x4 x 16ch pooled result.
//
// K ordering is tap-major: kk = (kh*3+kw)*CIN + cin (A and B agree, so the GEMM
// sum is unchanged).  Per tap, (iy,ix) and the padding predicate are invariant
// and the inner cin loop is pure pointer-increment:
//   A += 4*H*W floats, B += 4*16 floats per WMMA.  No muls in the hot loop.
//
// A layout (32-bit 16x4): lanes 0-15 VGPR{0,1}=K{0,1}; lanes 16-31 VGPR{0,1}=K{2,3}.
// D layout: VGPR r, lanes 0-15 -> (M=r, N=lane); lanes 16-31 -> (M=8+r, N=lane-16).
// => vertical pool partner = lane^16 (same VGPR), horizontal = VGPR pair (2q,2q+1).
template<int CIN, int COUT, int HIN, int WIN>
__global__ __launch_bounds__(256)
void conv3x3_lrelu_pool2(const float* __restrict__ in,    // [B, CIN, HIN, WIN]
                         const float* __restrict__ wrep,  // repacked [COUT/16][K][16]
                         const float* __restrict__ bias,  // [COUT]
                         float* __restrict__ out,         // [B, COUT, HIN/2, WIN/2]
                         float bscale, int B)
{
    constexpr int K       = CIN * 9;
    constexpr int HW      = HIN * WIN;
    constexpr int HP      = HIN / 2, WP = WIN / 2;
    constexpr int TILES_X = WIN / 8;   // 8 conv-out columns per tile
    constexpr int TILES_Y = HIN / 2;   // 2 conv-out rows per tile
    constexpr int NG      = COUT / 16;

    const int lane = threadIdx.x & 31;
    const int wid  = blockIdx.x * (blockDim.x >> 5) + (threadIdx.x >> 5);

    int t = wid;
    const int tx8 = t % TILES_X; t /= TILES_X;
    const int ty2 = t % TILES_Y; t /= TILES_Y;
    const int ng  = t % NG;      t /= NG;
    const int b   = t;
    if (b >= B) return;

    const int half2 = (lane >> 4) * 2;  // K-pair selector {0,2}
    const int m     = lane & 15;        // A row (pixel) == B/D column (channel)
    const int tyy   = m >> 3;           // conv-out row within tile (0..1)
    const int txx   = m & 7;            // conv-out col within tile (0..7)
    const int oy    = ty2 * 2 + tyy;
    const int ox    = tx8 * 8 + txx;
    const int n0    = ng * 16;

    const float* inp = in + (size_t)b * CIN * HW;
    const float* wg  = wrep + (size_t)ng * K * 16;

    v8f acc;
    const float bv = bias[n0 + m] * bscale;
#pragma unroll
    for (int i = 0; i < 8; ++i) acc[i] = bv;

#pragma unroll
    for (int p = 0; p < 9; ++p) {       // filter tap, compile-time
        const int kh = p / 3, kw = p % 3;
        const int iy = oy + kh - 1;
        const int ix = ox + kw - 1;
        const bool ok = (iy >= 0) & (iy < HIN) & (ix >= 0) & (ix < WIN);
        // clamped base so loads are always in-bounds; zero-padded via select
        const float* ap = inp + (size_t)((ok ? iy : 0) * WIN + (ok ? ix : 0))
                              + (size_t)half2 * HW;
        const float* bp = wg + (size_t)(p * CIN + half2) * 16 + m;
#pragma unroll 4
        for (int c0 = 0; c0 < CIN; c0 += 4) {
            v2f a, bm;
            const float a0 = ap[0];
            const float a1 = ap[HW];
            a[0]  = ok ? a0 : 0.0f;
            a[1]  = ok ? a1 : 0.0f;
            bm[0] = bp[0];      // coalesced: lanes share 16-float rows
            bm[1] = bp[16];
            // D = A x B + C  (f32, 16x16x4)
            acc = __builtin_amdgcn_wmma_f32_16x16x4_f32(false, a, false, bm,
                                                        (short)0, acc, false, false);
            ap += 4 * HW;
            bp += 4 * 16;
        }
    }

    // LeakyReLU(0.2): max(v, 0.2v) is exact for both signs
#pragma unroll
    for (int i = 0; i < 8; ++i)
        acc[i] = fmaxf(acc[i], 0.2f * acc[i]);

    // 2x2 maxpool fused in-register: vertical via lane^16, horizontal via VGPR pair
    float pq[4];
#pragma unroll
    for (int q = 0; q < 4; ++q) {
        const float v0 = acc[2 * q], v1 = acc[2 * q + 1];
        const float u0 = fmaxf(v0, __shfl_xor(v0, 16, 32));
        const float u1 = fmaxf(v1, __shfl_xor(v1, 16, 32));
        pq[q] = fmaxf(u0, u1);
    }

    if (half2 == 0) {
        // lane m == channel n0+m writes 4 consecutive pooled pixels (16B store)
        float* op = out + (((size_t)b * COUT + n0 + m) * HP + ty2) * WP + tx8 * 4;
        *(float4*)op = make_float4(pq[0], pq[1], pq[2], pq[3]);
    }
}

// out[b] = bc + (1/64) * sum_c sum_p u3[b][c][p] * wc[c]
__global__ __launch_bounds__(256)
void head_kernel(const float* __restrict__ u3,   // [B, 128, 8, 8]
                 const float* __restrict__ wc,   // [1, 128]
                 const float* __restrict__ bc,   // [1]
                 float* __restrict__ out, int B)
{
    const int b   = blockIdx.x;
    const int tid = threadIdx.x;
    if (b >= B) return;
    __shared__ float red[256];
    const float* p = u3 + (size_t)b * 128 * 64;
    float s = 0.0f;
    for (int i = tid; i < 128 * 64; i += 256)
        s += p[i] * wc[i >> 6];
    red[tid] = s;
    __syncthreads();
    for (int st = 128; st > 0; st >>= 1) {
        if (tid < st) red[tid] += red[tid + st];
        __syncthreads();
    }
    if (tid == 0) out[b] = red[0] * (1.0f / 64.0f) + bc[0];
}

extern "C" void kernel_launch(void* const* d_in, const int* in_sizes, int n_in,
                              void* d_out, int out_size, void* d_ws, size_t ws_size,
                              hipStream_t stream)
{
    const float* x  = (const float*)d_in[0];
    const float* w1 = (const float*)d_in[1];
    const float* b1 = (const float*)d_in[2];
    const float* w2 = (const float*)d_in[3];
    const float* b2 = (const float*)d_in[4];
    const float* w3 = (const float*)d_in[5];
    const float* b3 = (const float*)d_in[6];
    const float* wc = (const float*)d_in[7];
    const float* bc = (const float*)d_in[8];
    float* out = (float*)d_out;

    const int B = in_sizes[0] / (4 * 64 * 64);   // 32

    // Workspace: u1 | u2 | u3 | repacked weights  (~7.4MB total, L2-resident)
    float* u1  = (float*)d_ws;                          // [B,32,32,32]
    float* u2  = u1 + (size_t)B * 32 * 32 * 32;         // [B,64,16,16]
    float* u3  = u2 + (size_t)B * 64 * 16 * 16;         // [B,128,8,8]
    float* wr1 = u3 + (size_t)B * 128 * 8 * 8;          // 32*36    = 1152
    float* wr2 = wr1 + 32 * 36;                         // 64*288   = 18432
    float* wr3 = wr2 + 64 * 288;                        // 128*576  = 73728

    // Graph collapsed: node-0 output depends only on one conv chain with the
    // block-2 aggregation equal to 11*conv_w2 + 4*b2 (in-degrees 1+3+3+4).
    // Weight repack folds the x11 scale; x4 bias scale stays exact at C-init.
    repack_w<<<(32 * 36 + 255) / 256, 256, 0, stream>>>(w1, wr1, 32, 4, 1.0f);
    repack_w<<<(64 * 288 + 255) / 256, 256, 0, stream>>>(w2, wr2, 64, 32, 11.0f);
    repack_w<<<(128 * 576 + 255) / 256, 256, 0, stream>>>(w3, wr3, 128, 64, 1.0f);

    {   // layer 1: 4 -> 32, 64x64
        const int waves = B * (32 / 16) * (64 / 2) * (64 / 8);
        conv3x3_lrelu_pool2<4, 32, 64, 64>
            <<<(waves + 7) / 8, 256, 0, stream>>>(x, wr1, b1, u1, 1.0f, B);
    }
    {   // layer 2: 32 -> 64, 32x32, 11*conv + 4*b2
        const int waves = B * (64 / 16) * (32 / 2) * (32 / 8);
        conv3x3_lrelu_pool2<32, 64, 32, 32>
            <<<(waves + 7) / 8, 256, 0, stream>>>(u1, wr2, b2, u2, 4.0f, B);
    }
    {   // layer 3: 64 -> 128, 16x16
        const int waves = B * (128 / 16) * (16 / 2) * (16 / 8);
        conv3x3_lrelu_pool2<64, 128, 16, 16>
            <<<(waves + 7) / 8, 256, 0, stream>>>(u2, wr3, b3, u3, 1.0f, B);
    }
    head_kernel<<<B, 256, 0, stream>>>(u3, wc, bc, out, B);
}